// DFINEPostProcessor_75986561401445
// MI455X (gfx1250) — compile-verified
//
#include <hip/hip_runtime.h>
#include <math.h>

// Problem constants (from the reference).
#define NUM_B   512
#define NUM_Q   900
#define NUM_C   80
#define QC      (NUM_Q * NUM_C)   // 72000 scores per batch element
#define TOPK    300
#define KPT_F   34                // 17 keypoints * 2 floats
#define NBINS   2048              // 11-bit radix histogram
#define CAP     2048              // candidate buffer (power of two for bitonic)
#define TILE    1024              // floats per async tile (256 thr * float4)
#define THREADS 256               // 8 wave32 waves

// ---- CDNA5 async global->LDS path (guarded so it compiles either way) ----
#if __has_builtin(__builtin_amdgcn_global_load_async_to_lds_b128)
#define HAVE_ASYNC 1
#else
#define HAVE_ASYNC 0
#endif

// Exact pointee type from the compiler diagnostic: GCC-style int4 vector.
typedef int v4i __attribute__((vector_size(4 * sizeof(int))));
typedef __attribute__((address_space(1))) v4i gv4i;   // global
typedef __attribute__((address_space(3))) v4i lv4i;   // LDS

__device__ __forceinline__ void async_copy_b128(const float* g, float* l) {
#if HAVE_ASYNC
    __builtin_amdgcn_global_load_async_to_lds_b128(
        (gv4i*)(g), (lv4i*)(l), 0, 0);
#else
    *(float4*)l = *(const float4*)g;
#endif
}

#if HAVE_ASYNC
#if __has_builtin(__builtin_amdgcn_s_wait_asynccnt)
#define WAIT_ASYNC(n) __builtin_amdgcn_s_wait_asynccnt(n)
#else
#define WAIT_ASYNC(n) asm volatile("s_wait_asynccnt " #n ::: "memory")
#endif
#else
#define WAIT_ASYNC(n) ((void)0)
#endif

// Order-preserving float -> uint key (larger float => larger key).
__device__ __forceinline__ unsigned f2key(float f) {
    unsigned u = __float_as_uint(f);
    return (u & 0x80000000u) ? ~u : (u | 0x80000000u);
}
__device__ __forceinline__ float key2f(unsigned key) {
    unsigned u = (key & 0x80000000u) ? (key ^ 0x80000000u) : ~key;
    return __uint_as_float(u);
}

__global__ __launch_bounds__(THREADS)
void dfine_topk_kernel(const float* __restrict__ logits,
                       const float* __restrict__ boxes,
                       const float* __restrict__ kpts,
                       float* __restrict__ out) {
    const int b   = blockIdx.x;
    const int tid = threadIdx.x;
    const float* blog = logits + (size_t)b * QC;

    __shared__ unsigned hist[NBINS];
    __shared__ unsigned chunk[THREADS];
    __shared__ uint2    buf[CAP];
    __shared__ float    tiles[2][TILE];
    __shared__ unsigned s_cut;
    __shared__ unsigned s_cnt;

    for (int i = tid; i < NBINS; i += THREADS) hist[i] = 0u;
    if (tid == 0) s_cnt = 0u;
    __syncthreads();

    // ---------------- Pass 1: histogram via async double-buffered LDS tiles
    const int ntiles = (QC + TILE - 1) / TILE;   // 71 (last tile partial)
    {   // prefetch tile 0
        int g0 = tid * 4;
        if (g0 + 4 <= QC) async_copy_b128(blog + g0, &tiles[0][tid * 4]);
    }
    for (int it = 0; it < ntiles; ++it) {
        const int cur = it & 1;
        if (it + 1 < ntiles) {
            int gn = (it + 1) * TILE + tid * 4;
            if (gn + 4 <= QC) async_copy_b128(blog + gn, &tiles[cur ^ 1][tid * 4]);
            WAIT_ASYNC(1);      // oldest (current tile) complete, next in flight
        } else {
            WAIT_ASYNC(0);
        }
        __syncthreads();        // all waves' async writes to tiles[cur] visible

        const int gbase = it * TILE + tid * 4;
#pragma unroll
        for (int j = 0; j < 4; ++j) {
            int gi = gbase + j;
            if (gi < QC) {
                unsigned key = f2key(tiles[cur][tid * 4 + j]);
                atomicAdd(&hist[key >> 21], 1u);     // ds_add_u32
            }
        }
        __syncthreads();
    }

    // ---------------- Find cutoff bin: largest bin with suffix-count >= TOPK
    unsigned csum = 0;
#pragma unroll
    for (int j = 0; j < NBINS / THREADS; ++j) csum += hist[tid * (NBINS / THREADS) + j];
    chunk[tid] = csum;
    __syncthreads();
    if (tid == 0) {
        unsigned cum = 0, cut = 0;
        for (int c = THREADS - 1; c >= 0; --c) {
            if (cum + chunk[c] >= TOPK) {
                for (int bin = c * 8 + 7; bin >= c * 8; --bin) {
                    cum += hist[bin];
                    if (cum >= TOPK) { cut = (unsigned)bin; break; }
                }
                break;
            }
            cum += chunk[c];
        }
        s_cut = cut;
    }
    __syncthreads();
    const unsigned cut = s_cut;

    // ---------------- Pass 2 (L2-resident): collect candidates with bin >= cut
    for (int gi = tid * 4; gi < QC; gi += THREADS * 4) {
        float4 v = *(const float4*)(blog + gi);
        float vv[4] = {v.x, v.y, v.z, v.w};
#pragma unroll
        for (int j = 0; j < 4; ++j) {
            unsigned key = f2key(vv[j]);
            if ((key >> 21) >= cut) {
                unsigned pos = atomicAdd(&s_cnt, 1u);    // ds_add_rtn_u32
                if (pos < CAP) { uint2 e; e.x = key; e.y = (unsigned)(gi + j); buf[pos] = e; }
            }
        }
    }
    __syncthreads();
    unsigned m = s_cnt; if (m > CAP) m = CAP;
    for (unsigned i = m + tid; i < CAP; i += THREADS) {  // pad with -inf keys
        uint2 e; e.x = 0u; e.y = 0xFFFFFFFFu; buf[i] = e;
    }
    __syncthreads();

    // ---------------- Bitonic sort, descending by key, tie-break idx ascending
    for (unsigned k = 2; k <= CAP; k <<= 1) {
        for (unsigned j = k >> 1; j > 0; j >>= 1) {
            for (unsigned i = tid; i < CAP; i += THREADS) {
                unsigned l = i ^ j;
                if (l > i) {
                    uint2 a = buf[i], c = buf[l];
                    bool aLess    = (a.x < c.x) || (a.x == c.x && a.y > c.y);
                    bool aGreater = (a.x > c.x) || (a.x == c.x && a.y < c.y);
                    bool sw = ((i & k) == 0) ? aLess : aGreater;
                    if (sw) { buf[i] = c; buf[l] = a; }
                }
            }
            __syncthreads();
        }
    }

    // ---------------- Emit top-300: labels | boxes | scores | keypoints
    float* out_labels = out;                              // [B, T]
    float* out_boxes  = out + (size_t)NUM_B * TOPK;       // [B, T, 4]
    float* out_scores = out + (size_t)NUM_B * TOPK * 5;   // [B, T]
    float* out_kpts   = out + (size_t)NUM_B * TOPK * 6;   // [B, T, 17, 2]

    for (int e = tid; e < TOPK; e += THREADS) {
        uint2 ent = buf[e];
        float logit = key2f(ent.x);
        float score = 1.0f / (1.0f + __expf(-logit));
        unsigned idx = ent.y;
        unsigned q   = idx / NUM_C;
        unsigned lab = idx - q * NUM_C;
        size_t row = (size_t)b * TOPK + e;

        out_labels[row] = (float)lab;
        out_scores[row] = score;

        float4 bx = *(const float4*)(boxes + ((size_t)b * NUM_Q + q) * 4);
        *(float4*)(out_boxes + row * 4) = bx;

        const float2* kp = (const float2*)(kpts + ((size_t)b * NUM_Q + q) * KPT_F);
        float2* ko = (float2*)(out_kpts + row * KPT_F);
#pragma unroll
        for (int j = 0; j < KPT_F / 2; ++j) ko[j] = kp[j];
    }
}

extern "C" void kernel_launch(void* const* d_in, const int* in_sizes, int n_in,
                              void* d_out, int out_size, void* d_ws, size_t ws_size,
                              hipStream_t stream) {
    (void)in_sizes; (void)n_in; (void)out_size; (void)d_ws; (void)ws_size;
    const float* logits = (const float*)d_in[0];
    const float* boxes  = (const float*)d_in[1];
    const float* kpts   = (const float*)d_in[2];
    float* out = (float*)d_out;
    hipLaunchKernelGGL(dfine_topk_kernel, dim3(NUM_B), dim3(THREADS), 0, stream,
                       logits, boxes, kpts, out);
}